// EdgePruner_5325759447737
// MI455X (gfx1250) — compile-verified
//
#include <hip/hip_runtime.h>
#include <hip/hip_bf16.h>

// Problem constants (from reference): L=12, M=13, B=8, P=1024, H=12, Dh=64, Dm=768
#define NL 12
#define NM 13
#define NB 8
#define NP 1024
#define NH 12
#define NDH 64
#define NDM 768
#define KTOT (NH * NDH)        // 768 contiguous (h,d) per layer
// GEMM tiling
#define PT 128                 // p rows per block
#define NT 256                 // m cols per block
#define KC 32                  // K chunk staged in LDS (never crosses an h boundary)
#define LDA 36                 // padded LDS row stride for A (bank-conflict free)
#define LDW2 (2 * NT + 8)      // 520: padded stride for K-pair-interleaved W tile
#define THREADS 512            // 16 waves (wave32)

typedef __attribute__((ext_vector_type(2))) float v2f;
typedef __attribute__((ext_vector_type(4))) float v4f;
typedef __attribute__((ext_vector_type(8))) float v8f;

// ---------------- kernel 0: tiny bias precompute into workspace ----------------
// ws_pb[m]      = sum_l post_bias[l,m]
// ws_mlpb[b,m]  = sum_M (1 - mlp_mask[b,M]) * modal_mlp[M,m]
__global__ void precompute_bias(const float* __restrict__ post_bias,
                                const float* __restrict__ mlp_mask,
                                const float* __restrict__ modal_mlp,
                                float* __restrict__ ws_pb,
                                float* __restrict__ ws_mlpb) {
  int m = blockIdx.x * blockDim.x + threadIdx.x;       // grid.x = 3, 256 thr -> m in [0,768)
  if (m >= NDM) return;
  if (blockIdx.y == NB) {
    float s = 0.f;
    for (int l = 0; l < NL; ++l) s += post_bias[l * NDM + m];
    ws_pb[m] = s;
  } else {
    int b = blockIdx.y;
    float s = 0.f;
    for (int mm = 0; mm < NM; ++mm)
      s += (1.0f - mlp_mask[b * NM + mm]) * modal_mlp[mm * NDM + m];
    ws_mlpb[b * NDM + m] = s;
  }
}

// ---------------- kernel 1: streaming MLP reduction + biases (writes out) ------
// out[b,p,m] = sum_M mask[b,M]*mlp_cache[M,b,p,m] + ws_mlpb[b,m] + ws_pb[m]
__global__ __launch_bounds__(256) void mlp_stream(const float* __restrict__ mlp_cache,
                                                  const float* __restrict__ mlp_mask,
                                                  const float* __restrict__ ws_pb,
                                                  const float* __restrict__ ws_mlpb,
                                                  float* __restrict__ out) {
  // one float4 per thread; total float4 = B*P*(Dm/4) = 1,572,864
  int g = blockIdx.x * blockDim.x + threadIdx.x;
  const int NM4 = NDM / 4;                              // 192
  int m4 = g % NM4;
  int bp = g / NM4;                                     // b*P + p
  int b = bp >> 10;                                     // P = 1024
  v4f acc = *(const v4f*)(ws_pb + m4 * 4) + *(const v4f*)(ws_mlpb + b * NDM + m4 * 4);
  const float* mk = mlp_mask + b * NM;
  const float* base = mlp_cache + (long long)bp * NDM + m4 * 4;
  #pragma unroll
  for (int mm = 0; mm < NM; ++mm) {
    // mlp_cache flat index: ((mm*B + b)*P + p)*Dm + m  = (mm*B*P*Dm) + bp*Dm + m
    v4f c = *(const v4f*)(base + (long long)mm * NB * NP * NDM);
    acc += mk[mm] * c;
  }
  *(v4f*)(out + (long long)bp * NDM + m4 * 4) = acc;
}

// ---------------- kernel 2: attention GEMM via V_WMMA_F32_16X16X4_F32 ----------
// out[b,p,m] += sum_{l,h,d} (am[b,l,h]*attn_cache[l,b,p,h,d]
//                            + (1-am[b,l,h])*modal_attention[l,h,d]) * W_O[l,h,d,m]
__global__ __launch_bounds__(THREADS) void attn_gemm(const float* __restrict__ attn_cache,
                                                     const float* __restrict__ attn_mask,
                                                     const float* __restrict__ modal_attn,
                                                     const float* __restrict__ W_O,
                                                     float* __restrict__ out) {
  __shared__ __align__(16) float lds_A[PT * LDA];         // 128 x 32 (+pad) staged A
  __shared__ __align__(16) float lds_W[(KC / 2) * LDW2];  // 16 pair-rows x 256 cols x 2

  const int b  = blockIdx.z;
  const int p0 = blockIdx.y * PT;
  const int m0 = blockIdx.x * NT;
  const int tid  = threadIdx.x;
  const int lane = tid & 31;                            // wave32
  const int wave = tid >> 5;                            // 0..15
  const int half = lane >> 4;                           // 0/1 (lane half)
  const int lr   = lane & 15;
  const int wp   = wave >> 1;                           // 0..7  : 16-row subtile
  const int wn   = wave & 1;                            // 0..1  : 128-col subtile

  v8f acc[8];
  const v8f vzero = {0.f, 0.f, 0.f, 0.f, 0.f, 0.f, 0.f, 0.f};
  #pragma unroll
  for (int j = 0; j < 8; ++j) acc[j] = vzero;

  for (int l = 0; l < NL; ++l) {
    const float* Abase  = attn_cache + ((long long)(l * NB + b) * NP + p0) * KTOT;
    const float* Wbase  = W_O + (long long)l * KTOT * NDM + m0;
    const float* Mbase  = modal_attn + l * KTOT;
    const float* ambase = attn_mask + (b * NL + l) * NH;

    for (int kc = 0; kc < KTOT; kc += KC) {
      __syncthreads();                                  // prior-chunk LDS reads done

      // --- stage A (128 x 32), fold in mask scale + modal bias ---
      // h is constant across the 32-wide chunk (Dh=64, kc multiple of 32)
      const float s  = ambase[kc >> 6];
      const float t1 = 1.0f - s;
      {
        const int q  = tid & 7;                         // float4 column group 0..7
        const int pr = tid >> 3;                        // 0..63
        const v4f mo = *(const v4f*)(Mbase + kc + q * 4);
        #pragma unroll
        for (int it = 0; it < 2; ++it) {
          const int p = pr + it * 64;                   // 0..127
          v4f c = *(const v4f*)(Abase + (long long)p * KTOT + kc + q * 4);
          *(v4f*)(lds_A + p * LDA + q * 4) = s * c + t1 * mo;
        }
      }
      // --- stage W (32 x 256) in K-pair-interleaved layout: lds_W[k/2][col][2] ---
      // so a B fragment {W[k][col], W[k+1][col]} is one aligned ds_load_b64.
      #pragma unroll
      for (int it = 0; it < 2; ++it) {
        const int f  = tid + it * THREADS;              // 0..1023
        const int pr = f >> 6;                          // K pair-row 0..15
        const int cg = f & 63;                          // 4-col group 0..63
        const int k  = 2 * pr;
        v4f r0 = *(const v4f*)(Wbase + (long long)(kc + k) * NDM + cg * 4);      // row k
        v4f r1 = *(const v4f*)(Wbase + (long long)(kc + k + 1) * NDM + cg * 4);  // row k+1
        float* dst = lds_W + pr * LDW2 + cg * 8;
        v4f lo = {r0.x, r1.x, r0.y, r1.y};              // cols cg*4, cg*4+1
        v4f hi = {r0.z, r1.z, r0.w, r1.w};              // cols cg*4+2, cg*4+3
        *(v4f*)(dst)     = lo;
        *(v4f*)(dst + 4) = hi;
      }
      __syncthreads();

      // --- compute: 8 K-steps of 4, 8 N-subtiles each ---
      const float* Ap = lds_A + (wp * 16 + lr) * LDA;
      #pragma unroll
      for (int kk = 0; kk < KC; kk += 4) {
        v2f a;
        a.x = Ap[kk + 2 * half];                        // A 16x4: lanes0-15 K0/K1, lanes16-31 K2/K3
        a.y = Ap[kk + 2 * half + 1];
        // B pair-row for this lane half: (kk + 2*half)/2
        const float* W0 = lds_W + ((kk >> 1) + half) * LDW2 + (wn * 128 + lr) * 2;
        #pragma unroll
        for (int j = 0; j < 8; ++j) {
          v2f bb = *(const v2f*)(W0 + j * 32);          // {W[k][col], W[k+1][col]}
          acc[j] = __builtin_amdgcn_wmma_f32_16x16x4_f32(
              false, a, false, bb, (short)0, acc[j], false, false);
        }
      }
    }
  }

  // --- epilogue: out += acc  (C layout: VGPR i -> M = i + 8*half, N = lr) ---
  float* Obase = out + ((long long)(b * NP + p0 + wp * 16 + 8 * half)) * NDM
                     + m0 + wn * 128 + lr;
  #pragma unroll
  for (int j = 0; j < 8; ++j) {
    float* Oj = Obase + j * 16;
    #pragma unroll
    for (int i = 0; i < 8; ++i) {
      Oj[(long long)i * NDM] += acc[j][i];
    }
  }
}

// ---------------- kernel 3: overwrite p=0 row with orig_in ----------------------
__global__ void copy_row0(const float* __restrict__ orig_in, float* __restrict__ out) {
  int m = blockIdx.x * blockDim.x + threadIdx.x;        // grid.x=3, 256 thr
  int b = blockIdx.y;
  if (m >= NDM) return;
  long long idx = (long long)(b * NP) * NDM + m;        // p = 0
  out[idx] = orig_in[idx];
}

extern "C" void kernel_launch(void* const* d_in, const int* in_sizes, int n_in,
                              void* d_out, int out_size, void* d_ws, size_t ws_size,
                              hipStream_t stream) {
  const float* attn_cache      = (const float*)d_in[0];
  const float* mlp_cache       = (const float*)d_in[1];
  const float* mlp_mask        = (const float*)d_in[2];
  const float* attn_mask       = (const float*)d_in[3];
  const float* modal_attention = (const float*)d_in[4];
  const float* modal_mlp       = (const float*)d_in[5];
  const float* W_O             = (const float*)d_in[6];
  const float* post_bias       = (const float*)d_in[7];
  const float* orig_in         = (const float*)d_in[8];
  float* out = (float*)d_out;

  float* ws_pb   = (float*)d_ws;        // 768 floats
  float* ws_mlpb = ws_pb + NDM;         // 8*768 floats

  // 0) tiny bias precompute
  precompute_bias<<<dim3(3, NB + 1), 256, 0, stream>>>(post_bias, mlp_mask, modal_mlp,
                                                       ws_pb, ws_mlpb);
  // 1) streaming MLP reduction + biases -> writes out
  {
    const int total_f4 = NB * NP * (NDM / 4);           // 1,572,864
    mlp_stream<<<total_f4 / 256, 256, 0, stream>>>(mlp_cache, mlp_mask, ws_pb, ws_mlpb, out);
  }
  // 2) WMMA f32 GEMM accumulates attention term into out
  attn_gemm<<<dim3(NDM / NT, NP / PT, NB), THREADS, 0, stream>>>(attn_cache, attn_mask,
                                                                 modal_attention, W_O, out);
  // 3) restore row p=0 from orig_in
  copy_row0<<<dim3(3, NB), 256, 0, stream>>>(orig_in, out);
}